// GNN_EBM_59493886984696
// MI455X (gfx1250) — compile-verified
//
#include <hip/hip_runtime.h>

// CDNA5 / gfx1250: wave32, WMMA f32 16x16x4 (A,B = 2 VGPRs each, C/D = 8 VGPRs)
typedef __attribute__((ext_vector_type(2))) float v2f;
typedef __attribute__((ext_vector_type(8))) float v8f;

#define D_X     100
#define D_NODES 102
#define D_IN    102      // D_X + 1 + D_Y
#define HID     256
#define MLP_HID 128
#define KP      104      // D_IN padded to multiple of 4
#define TM      16       // batch rows per block

__global__ __launch_bounds__(256) void gnn_ebm_fused(
    const float* __restrict__ x,     const float* __restrict__ t,
    const float* __restrict__ y,     const float* __restrict__ Bp,
    const float* __restrict__ fc_w,  const float* __restrict__ fc_b,
    const float* __restrict__ eT_w1, const float* __restrict__ eT_b1,
    const float* __restrict__ eT_w2, const float* __restrict__ eT_b2,
    const float* __restrict__ eY_w1, const float* __restrict__ eY_b1,
    const float* __restrict__ eY_w2, const float* __restrict__ eY_b2,
    float* __restrict__ out)
{
    __shared__ float zs[TM][KP];      // z tile (x|t|y), zero-padded K
    __shared__ float rs[TM][HID];     // relu(h0) tile
    __shared__ float ms[TM][HID];     // [eT hidden | eY hidden] pre-activation
    __shared__ float cs[D_NODES];     // conv1 per-node scale c_j
    __shared__ float ab[2];           // alpha, beta
    __shared__ float red[256];

    const int tid  = threadIdx.x;
    const int lane = tid & 31;
    const int wave = tid >> 5;
    const int blk  = blockIdx.x;
    const float inv_n = 1.0f / (float)D_NODES;

    // ---- Step 0: collapse the 2-layer GNN to scalars alpha (node 100), beta (node 101)
    if (tid < D_NODES) {
        const int j = tid;
        float s = 0.f;
        for (int k = 0; k < D_NODES; ++k) {
            float mk = 1.f;
            if (k == j) mk = 0.f;
            if (j == D_NODES - 1 && k < D_X) mk = 0.f;
            s += mk / (1.f + __expf(-Bp[j * D_NODES + k]));
        }
        cs[j] = 1.f + inv_n * s;
    }
    __syncthreads();
    if (tid < 2) {
        const int i = D_NODES - 2 + tid;   // 100 or 101
        float s = 0.f;
        for (int j = 0; j < D_NODES; ++j) {
            float mj = 1.f;
            if (j == i) mj = 0.f;
            if (i == D_NODES - 1 && j < D_X) mj = 0.f;
            s += (mj / (1.f + __expf(-Bp[i * D_NODES + j]))) * cs[j];
        }
        ab[tid] = cs[i] + inv_n * s;
    }

    // ---- Stage z = [x | t | y] tile into LDS (pad K to 104 with zeros)
    for (int idx = tid; idx < TM * KP; idx += 256) {
        const int rrow = idx / KP, cc = idx % KP;
        const int grow = blk * TM + rrow;
        float v;
        if (cc < D_X)           v = x[grow * D_X + cc];
        else if (cc == D_X)     v = t[grow];
        else if (cc == D_X + 1) v = y[grow];
        else                    v = 0.f;
        zs[rrow][cc] = v;
    }
    __syncthreads();

    // WMMA f32 16x16x4 fragment coordinates (wave32, ISA 7.12.2)
    const int mrow  = lane & 15;            // A: row M
    const int khalf = (lane >> 4) << 1;     // A/B: K base 0 or 2 within step
    const int ncol  = lane & 15;            // B/D: column N

    // ---- GEMM1: r = relu(z @ fc_in_w^T + fc_in_b)   [16 x 256], K = 104
    for (int nt = wave; nt < HID / 16; nt += 8) {
        v8f acc = {};
        const int n = nt * 16 + ncol;
        for (int ks = 0; ks < KP / 4; ++ks) {
            const int k0 = ks * 4 + khalf;
            v2f a, b;
            a.x = zs[mrow][k0];
            a.y = zs[mrow][k0 + 1];
            b.x = (k0     < D_IN) ? fc_w[n * D_IN + k0]     : 0.f;
            b.y = (k0 + 1 < D_IN) ? fc_w[n * D_IN + k0 + 1] : 0.f;
            acc = __builtin_amdgcn_wmma_f32_16x16x4_f32(
                      false, a, false, b, (short)0, acc, false, false);
        }
        const float bias = fc_b[n];
        for (int rr = 0; rr < 8; ++rr) {
            const int mr = rr + (lane >> 4) * 8;
            const float v = acc[rr] + bias;
            rs[mr][n] = v > 0.f ? v : 0.f;
        }
    }
    __syncthreads();

    // ---- GEMM2: m = r @ [eT_w1 ; eY_w1]^T   [16 x 256], K = 256
    for (int nt = wave; nt < HID / 16; nt += 8) {
        v8f acc = {};
        const int n = nt * 16 + ncol;
        const float* Wrow = (n < MLP_HID) ? (eT_w1 + n * HID)
                                          : (eY_w1 + (n - MLP_HID) * HID);
        for (int ks = 0; ks < HID / 4; ++ks) {
            const int k0 = ks * 4 + khalf;
            v2f a, b;
            a.x = rs[mrow][k0];
            a.y = rs[mrow][k0 + 1];
            b.x = Wrow[k0];
            b.y = Wrow[k0 + 1];
            acc = __builtin_amdgcn_wmma_f32_16x16x4_f32(
                      false, a, false, b, (short)0, acc, false, false);
        }
        for (int rr = 0; rr < 8; ++rr) {
            const int mr = rr + (lane >> 4) * 8;
            ms[mr][n] = acc[rr];
        }
    }
    __syncthreads();

    // ---- Epilogue: e = relu(a*m_T + b1T)@w2T + b2T + relu(b*m_Y + b1Y)@w2Y + b2Y
    const float alpha = ab[0], beta = ab[1];
    {
        const int row = tid >> 4;
        const int c0  = tid & 15;
        float partial = 0.f;
        for (int c = c0; c < MLP_HID; c += 16) {
            float vT = alpha * ms[row][c] + eT_b1[c];
            vT = vT > 0.f ? vT : 0.f;
            float vY = beta * ms[row][MLP_HID + c] + eY_b1[c];
            vY = vY > 0.f ? vY : 0.f;
            partial += vT * eT_w2[c] + vY * eY_w2[c];
        }
        red[tid] = partial;
    }
    __syncthreads();
    if ((tid & 15) == 0) {
        const int row = tid >> 4;
        float s = 0.f;
        for (int i = 0; i < 16; ++i) s += red[row * 16 + i];
        out[blk * TM + row] = s + eT_b2[0] + eY_b2[0];
    }
}

extern "C" void kernel_launch(void* const* d_in, const int* in_sizes, int n_in,
                              void* d_out, int out_size, void* d_ws, size_t ws_size,
                              hipStream_t stream) {
    const float* x     = (const float*)d_in[0];
    const float* t     = (const float*)d_in[1];
    const float* y     = (const float*)d_in[2];
    const float* Bp    = (const float*)d_in[3];
    const float* fc_w  = (const float*)d_in[4];
    const float* fc_b  = (const float*)d_in[5];
    const float* eT_w1 = (const float*)d_in[6];
    const float* eT_b1 = (const float*)d_in[7];
    const float* eT_w2 = (const float*)d_in[8];
    const float* eT_b2 = (const float*)d_in[9];
    const float* eY_w1 = (const float*)d_in[10];
    const float* eY_b1 = (const float*)d_in[11];
    const float* eY_w2 = (const float*)d_in[12];
    const float* eY_b2 = (const float*)d_in[13];
    float* out = (float*)d_out;

    const int batch = in_sizes[1];         // t has BATCH elements
    const int nblocks = batch / TM;        // 2048/16 = 128

    gnn_ebm_fused<<<nblocks, 256, 0, stream>>>(
        x, t, y, Bp, fc_w, fc_b,
        eT_w1, eT_b1, eT_w2, eT_b2,
        eY_w1, eY_b1, eY_w2, eY_b2, out);
}